// FreqCompression_81930796138567
// MI455X (gfx1250) — compile-verified
//
#include <hip/hip_runtime.h>

#define NFREQ 257
#define NFILT 64
#define IN_C  64
#define HID   128
#define TLEN  1000
#define BATCH 2

#define MT       64            // t rows per workgroup
#define MAXLSTR  34            // generic kernel: supports jpad up to 32

typedef float v2f __attribute__((ext_vector_type(2)));
typedef float v8f __attribute__((ext_vector_type(8)));

// ---------------------------------------------------------------------------
// Specialized kernel: JMAX known at compile time -> fully unrolled K loop
// (24 static v_wmma per channel), double-buffered LDS, software-pipelined
// global loads. Grid = (t-tiles, filter, batch) so b is uniform.
// ---------------------------------------------------------------------------
template <int JMAXC>
__global__ __launch_bounds__(256)
void freqcomp_wmma_f32_t(const float* __restrict__ x,
                         const float* __restrict__ W,
                         const int*   __restrict__ idx,
                         float*       __restrict__ out)
{
    constexpr int JPAD = (JMAXC + 3) & ~3;   // 24 for JMAXC=21
    constexpr int LSTR = JPAD + 2;           // 26: even stride, 8B-aligned b64 LDS frags
    constexpr int NA   = JPAD / 4;           // A elems per thread (6)
    constexpr int NB   = JPAD / 2;           // B elems per thread (12)

    __shared__ __align__(16) float lA[2][MT  * LSTR];
    __shared__ __align__(16) float lB[2][HID * LSTR];

    const int tbase = blockIdx.x * MT;
    const int f     = blockIdx.y;
    const int b     = blockIdx.z;
    const int start = idx[f * JMAXC];        // idx[f,j] == clip(start + j, 0, NFREQ-1)

    const int tid  = threadIdx.x;
    const int lane = tid & 31;               // wave32
    const int wave = tid >> 5;               // 0..7
    const int wm   = wave & 3;               // 16-row M sub-tile
    const int wn   = wave >> 2;              // 64-col N half

    // staging maps
    const int arow = tid >> 2;               // 0..63  (4 threads per A row)
    const int aq   = tid & 3;
    const int brow = tid >> 1;               // 0..127 (2 threads per B row)
    const int bq   = tid & 1;

    int trow = tbase + arow;
    trow = trow < TLEN ? trow : TLEN - 1;    // clamp loads; stores masked later
    const float* xrow = x + (((size_t)b * IN_C) * TLEN + trow) * NFREQ;
    const float* wrow = W + ((size_t)(f * HID + brow) * IN_C) * JMAXC;

    float ar[NA];
    float br[NB];

    auto gload = [&](int c) {
        const float* rp = xrow + (size_t)c * TLEN * NFREQ;
        #pragma unroll
        for (int i = 0; i < NA; ++i) {
            int col = start + aq * NA + i;
            col = col < NFREQ ? col : NFREQ - 1;
            ar[i] = rp[col];
        }
        const float* wp = wrow + c * JMAXC;
        #pragma unroll
        for (int i = 0; i < NB; ++i) {
            int j = bq * NB + i;
            br[i] = (j < JMAXC) ? wp[j] : 0.0f;   // predicated load; zero pad cols
        }
    };
    auto stlds = [&](int buf) {
        #pragma unroll
        for (int i = 0; i < NA; ++i) lA[buf][arow * LSTR + aq * NA + i] = ar[i];
        #pragma unroll
        for (int i = 0; i < NB; ++i) lB[buf][brow * LSTR + bq * NB + i] = br[i];
    };

    v8f acc0 = {}, acc1 = {}, acc2 = {}, acc3 = {};

    const int ml   = lane & 15;
    const int kk   = (lane >> 4) << 1;       // K pair within a 4-wide step
    const int aoff = (wm * 16 + ml) * LSTR + kk;
    const int boff = (wn * 64 + ml) * LSTR + kk;

    gload(0);
    int buf = 0;
    for (int c = 0; c < IN_C; ++c) {
        stlds(buf);
        __syncthreads();
        if (c + 1 < IN_C) gload(c + 1);      // HBM loads fly while we do WMMA below

        const float* pa = &lA[buf][aoff];
        const float* pb = &lB[buf][boff];
        #pragma unroll
        for (int ks = 0; ks < JPAD; ks += 4) {
            v2f a  = *(const v2f*)(pa + ks);
            v2f b0 = *(const v2f*)(pb + ks);
            v2f b1 = *(const v2f*)(pb + 16 * LSTR + ks);
            v2f b2 = *(const v2f*)(pb + 32 * LSTR + ks);
            v2f b3 = *(const v2f*)(pb + 48 * LSTR + ks);
            acc0 = __builtin_amdgcn_wmma_f32_16x16x4_f32(false, a, false, b0, (short)0, acc0, false, false);
            acc1 = __builtin_amdgcn_wmma_f32_16x16x4_f32(false, a, false, b1, (short)0, acc1, false, false);
            acc2 = __builtin_amdgcn_wmma_f32_16x16x4_f32(false, a, false, b2, (short)0, acc2, false, false);
            acc3 = __builtin_amdgcn_wmma_f32_16x16x4_f32(false, a, false, b3, (short)0, acc3, false, false);
        }
        buf ^= 1;
        // one barrier per channel: buf is only rewritten two iterations after
        // its last read, and every wave barriers after finishing its reads.
    }

    // epilogue: D layout — VGPR r: M(t) = r + 8*(lane>=16), N(h) = lane&15
    const int t0    = tbase + wm * 16 + ((lane >> 4) << 3);  // lane's first t
    const int hbase = wn * 64 + ml;
    float* op = out + (((size_t)b * HID + hbase) * TLEN + t0) * NFILT + f;
    constexpr size_t HSTEP = (size_t)16 * TLEN * NFILT;      // +16 h rows

    if (tbase + MT <= TLEN) {
        // fast path (uniform): whole tile in bounds, no masks at all
        #pragma unroll
        for (int s = 0; s < 4; ++s) {
            v8f a = s == 0 ? acc0 : s == 1 ? acc1 : s == 2 ? acc2 : acc3;
            float* p = op + (size_t)s * HSTEP;
            #pragma unroll
            for (int r = 0; r < 8; ++r) p[(size_t)r * NFILT] = a[r];
        }
    } else {
        #pragma unroll
        for (int s = 0; s < 4; ++s) {
            v8f a = s == 0 ? acc0 : s == 1 ? acc1 : s == 2 ? acc2 : acc3;
            float* p = op + (size_t)s * HSTEP;
            #pragma unroll
            for (int r = 0; r < 8; ++r)
                if (t0 + r < TLEN) p[(size_t)r * NFILT] = a[r];
        }
    }
}

// ---------------------------------------------------------------------------
// Generic fallback (runtime jmax), same algorithm, single-buffered.
// ---------------------------------------------------------------------------
__global__ __launch_bounds__(256)
void freqcomp_wmma_f32_g(const float* __restrict__ x,
                         const float* __restrict__ W,
                         const int*   __restrict__ idx,
                         float*       __restrict__ out,
                         int jmax)
{
    __shared__ __align__(16) float lA[MT  * MAXLSTR];
    __shared__ __align__(16) float lB[HID * MAXLSTR];

    const int jpad = (jmax + 3) & ~3;
    const int lstr = jpad + 2;

    const int tbase = blockIdx.x * MT;
    const int f     = blockIdx.y;
    const int b     = blockIdx.z;
    const int start = idx[f * jmax];

    const int tid  = threadIdx.x;
    const int lane = tid & 31;
    const int wave = tid >> 5;
    const int wm   = wave & 3;
    const int wn   = wave >> 2;

    v8f acc0 = {}, acc1 = {}, acc2 = {}, acc3 = {};

    const int arow = tid >> 2;
    const int aq   = tid & 3;
    const int brow = tid >> 1;
    const int bq   = tid & 1;

    int trow = tbase + arow;
    trow = trow < TLEN ? trow : TLEN - 1;
    const float* xrow = x + (((size_t)b * IN_C) * TLEN + trow) * NFREQ;

    const int ml = lane & 15;
    const int kk = (lane >> 4) << 1;
    const float* pa  = &lA[(wm * 16 + ml) * lstr + kk];
    const float* pb0 = &lB[(wn * 64 + ml) * lstr + kk];

    for (int c = 0; c < IN_C; ++c) {
        {
            const float* rp = xrow + (size_t)c * TLEN * NFREQ;
            const int jn = jpad >> 2;
            for (int i = 0; i < jn; ++i) {
                int j   = aq * jn + i;
                int col = start + j;
                col = col < NFREQ ? col : NFREQ - 1;
                lA[arow * lstr + j] = rp[col];
            }
        }
        {
            const float* wp = W + (((size_t)f * HID + brow) * IN_C + c) * jmax;
            const int jn = jpad >> 1;
            for (int i = 0; i < jn; ++i) {
                int j = bq * jn + i;
                lB[brow * lstr + j] = (j < jmax) ? wp[j] : 0.0f;
            }
        }
        __syncthreads();
        for (int ks = 0; ks < jpad; ks += 4) {
            v2f a  = *(const v2f*)(pa  + ks);
            v2f b0 = *(const v2f*)(pb0 + ks);
            v2f b1 = *(const v2f*)(pb0 + 16 * lstr + ks);
            v2f b2 = *(const v2f*)(pb0 + 32 * lstr + ks);
            v2f b3 = *(const v2f*)(pb0 + 48 * lstr + ks);
            acc0 = __builtin_amdgcn_wmma_f32_16x16x4_f32(false, a, false, b0, (short)0, acc0, false, false);
            acc1 = __builtin_amdgcn_wmma_f32_16x16x4_f32(false, a, false, b1, (short)0, acc1, false, false);
            acc2 = __builtin_amdgcn_wmma_f32_16x16x4_f32(false, a, false, b2, (short)0, acc2, false, false);
            acc3 = __builtin_amdgcn_wmma_f32_16x16x4_f32(false, a, false, b3, (short)0, acc3, false, false);
        }
        __syncthreads();
    }

    const int t0    = tbase + wm * 16 + ((lane >> 4) << 3);
    const int hbase = wn * 64 + ml;
    float* op = out + (((size_t)b * HID + hbase) * TLEN + t0) * NFILT + f;
    const size_t HSTEP = (size_t)16 * TLEN * NFILT;

    if (tbase + MT <= TLEN) {
        #pragma unroll
        for (int s = 0; s < 4; ++s) {
            v8f a = s == 0 ? acc0 : s == 1 ? acc1 : s == 2 ? acc2 : acc3;
            float* p = op + (size_t)s * HSTEP;
            #pragma unroll
            for (int r = 0; r < 8; ++r) p[(size_t)r * NFILT] = a[r];
        }
    } else {
        #pragma unroll
        for (int s = 0; s < 4; ++s) {
            v8f a = s == 0 ? acc0 : s == 1 ? acc1 : s == 2 ? acc2 : acc3;
            float* p = op + (size_t)s * HSTEP;
            #pragma unroll
            for (int r = 0; r < 8; ++r)
                if (t0 + r < TLEN) p[(size_t)r * NFILT] = a[r];
        }
    }
}

extern "C" void kernel_launch(void* const* d_in, const int* in_sizes, int n_in,
                              void* d_out, int out_size, void* d_ws, size_t ws_size,
                              hipStream_t stream)
{
    const float* x   = (const float*)d_in[0];
    const float* W   = (const float*)d_in[1];
    const int*   idx = (const int*)d_in[2];
    float* out = (float*)d_out;

    const int jmax = in_sizes[1] / (NFILT * HID * IN_C);   // W = (F, H, C, jmax)

    dim3 grid((TLEN + MT - 1) / MT, NFILT, BATCH);         // 16 x 64 x 2 workgroups
    if (jmax == 21) {
        freqcomp_wmma_f32_t<21><<<grid, 256, 0, stream>>>(x, W, idx, out);
    } else {
        freqcomp_wmma_f32_g<<<grid, 256, 0, stream>>>(x, W, idx, out, jmax);
    }
}